// PillarMaxPooling_223338300085
// MI455X (gfx1250) — compile-verified
//
#include <hip/hip_runtime.h>

typedef float v2f __attribute__((ext_vector_type(2)));
typedef float v8f __attribute__((ext_vector_type(8)));

#define C_IN   10
#define KPAIRS 6        // K padded 10 -> 12 (row 10 = BN bias, row 11 = 0)
#define C_OUT  64
#define BN_EPS 1e-3f

// relu(x) = med3(x, 0, +inf): single v_med3_f32, no canonicalize pair.
__device__ __forceinline__ float relu1(float x) {
    return __builtin_amdgcn_fmed3f(x, 0.0f, __builtin_inff());
}

// ---------------------------------------------------------------------------
// Zero-fill output with 128-bit stores (harness poisons d_out; 0 is the
// identity for the ReLU'd segment-max and matches the reference's clamp of
// empty pillars to 0). out_size = M*64 floats -> divisible by 4.
// ---------------------------------------------------------------------------
__global__ void pillar_zero_kernel(float4* __restrict__ out, long n4) {
    long i = (long)blockIdx.x * blockDim.x + threadIdx.x;
    if (i < n4) out[i] = make_float4(0.f, 0.f, 0.f, 0.f);
}

// ---------------------------------------------------------------------------
// Fused Linear(10->64, bias-free) + BN(eval) + ReLU + scatter-max.
// BN folded entirely into the WMMA: B rows pre-scaled by
// inv = gamma*rsqrt(var+eps), and K-row 10 carries the BN bias with A[:,10]=1.
// One wave32 = 16 points x 64 channels via 12x V_WMMA_F32_16X16X4_F32.
// Block = 256 threads = 8 waves = 128 points.
// ---------------------------------------------------------------------------
__global__ __launch_bounds__(256) void pillar_mlp_maxpool_wmma(
    const float* __restrict__ gf,      // (P, 10)
    const int*   __restrict__ pilIdx,  // (P,)
    const float* __restrict__ W,       // (10, 64)
    const float* __restrict__ gamma,
    const float* __restrict__ beta,
    const float* __restrict__ rmean,
    const float* __restrict__ rvar,
    float*       __restrict__ out,     // (M, 64), pre-zeroed
    int P)
{
    // sB[kp][n] = {Wfold[2kp][n], Wfold[2kp+1][n]} where Wfold rows 0..9 are
    // W*inv, row 10 is the BN bias, row 11 is 0. One ds_load_b64 per operand.
    __shared__ v2f sB[KPAIRS][C_OUT];

    const int t = threadIdx.x;

    {
        // i = t and i = t+256 share n = i & 63 (256 % 64 == 0): compute the
        // folded BN constants once per thread.
        const int n = t & 63;
        float inv  = gamma[n] * rsqrtf(rvar[n] + BN_EPS);
        float bias = beta[n] - rmean[n] * inv;
        for (int i = t; i < KPAIRS * C_OUT; i += 256) {
            int kp = i >> 6;
            v2f w;
            w.x = (2 * kp     < C_IN) ? W[(2 * kp)     * C_OUT + n] * inv : bias;
            w.y = (2 * kp + 1 < C_IN) ? W[(2 * kp + 1) * C_OUT + n] * inv : 0.0f;
            sB[kp][n] = w;   // kp==5: {bias, 0} = K rows 10,11
        }
    }
    __syncthreads();

    const int wave = t >> 5;
    const int lane = t & 31;
    const int nl   = lane & 15;      // column within a 16-wide tile
    const int hi   = lane >> 4;      // half-wave select

    const long pbase = ((long)blockIdx.x * 8 + wave) * 16;
    if (pbase >= P) return;          // uniform within wave

    // This lane's A-row point (M = nl); clamp tail loads in-bounds.
    long myPoint = pbase + nl;
    long ldPoint = myPoint < P ? myPoint : (long)P - 1;
    const float* grow = gf + ldPoint * C_IN;

    int myPillar = pilIdx[ldPoint];  // broadcast later via shfl

    v8f acc[4] = {};                 // 4 channel tiles of 16

    // K = 0..11 in three 16x16x4 fp32 WMMA steps; K-row 10 injects the bias.
    #pragma unroll
    for (int kc = 0; kc < 3; ++kc) {
        const int koff = kc * 4 + hi * 2;        // 0,2,4,6,8,10
        const int ks   = (koff > 8) ? 8 : koff;  // clamp: keep load in-bounds
        v2f a = *(const v2f*)(grow + ks);        // aligned float2 global load
        if (koff > 8) { a.x = 1.0f; a.y = 0.0f; }  // bias row: A[:,10]=1, A[:,11]=0

        const int kp = kc * 2 + hi;              // paired-B row in LDS
        #pragma unroll
        for (int n = 0; n < 4; ++n) {
            v2f b = sB[kp][n * 16 + nl];         // single ds_load_b64
            acc[n] = __builtin_amdgcn_wmma_f32_16x16x4_f32(
                false, a, false, b, (short)0, acc[n], false, false);
        }
    }

    // Epilogue: acc already holds h*inv + bias -> ReLU (v_med3) + scatter-max.
    // D layout: VGPR r -> M = r + 8*hi, N = nl (+16 per tile).
    // v >= 0 so u32 ordering == float ordering; out is pre-zeroed.
    if (pbase + 16 <= (long)P) {
        // Fast path (always taken when P % 16 == 0): 1 bpermute + 1 address
        // calc per row, 4 immediate-offset non-returning atomics.
        #pragma unroll
        for (int r = 0; r < 8; ++r) {
            int pp = __shfl(myPillar, r + 8 * hi, 32);
            unsigned int* base = (unsigned int*)out + (size_t)pp * C_OUT + nl;
            #pragma unroll
            for (int n = 0; n < 4; ++n)
                atomicMax(base + n * 16, __float_as_uint(relu1(acc[n][r])));
        }
    } else {
        #pragma unroll
        for (int r = 0; r < 8; ++r) {
            int m  = r + 8 * hi;
            int pp = __shfl(myPillar, m, 32);
            unsigned int* base = (unsigned int*)out + (size_t)pp * C_OUT + nl;
            if (pbase + m < (long)P) {
                #pragma unroll
                for (int n = 0; n < 4; ++n)
                    atomicMax(base + n * 16, __float_as_uint(relu1(acc[n][r])));
            }
        }
    }
}

// ---------------------------------------------------------------------------
extern "C" void kernel_launch(void* const* d_in, const int* in_sizes, int n_in,
                              void* d_out, int out_size, void* d_ws, size_t ws_size,
                              hipStream_t stream) {
    const float* gf     = (const float*)d_in[0];
    const int*   pilIdx = (const int*)  d_in[1];
    // d_in[2] = num_pillars (device scalar); implied by out_size/64.
    const float* W      = (const float*)d_in[3];
    const float* gamma  = (const float*)d_in[4];
    const float* beta   = (const float*)d_in[5];
    const float* rmean  = (const float*)d_in[6];
    const float* rvar   = (const float*)d_in[7];
    float* out = (float*)d_out;

    const int  P  = in_sizes[0] / C_IN;
    const long n4 = (long)out_size / 4;   // 128-bit zero stores

    int zb = (int)((n4 + 255) / 256);
    pillar_zero_kernel<<<zb, 256, 0, stream>>>((float4*)out, n4);

    int blocks = (P + 127) / 128;   // 8 waves/block, 16 points/wave
    pillar_mlp_maxpool_wmma<<<blocks, 256, 0, stream>>>(
        gf, pilIdx, W, gamma, beta, rmean, rvar, out, P);
}